// GNNEncoder_57475252355504
// MI455X (gfx1250) — compile-verified
//
#include <hip/hip_runtime.h>
#include <math.h>

#define GCNT 8
#define NPG0 16384
#define EDG  (8 * 16384 * 8)      // 1,048,576 edges
#define CDIM 128
#define HDIM 256
#define OUTD 128
#define NMAX (GCNT * NPG0)        // 131072 nodes at layer 0
#define MAXS 4096                 // score histogram bins (scores ~ Poisson(8))
#define EPS_FA 0.1f

typedef __attribute__((ext_vector_type(2))) float v2f;
typedef __attribute__((ext_vector_type(8))) float v8f;
typedef __attribute__((ext_vector_type(4))) unsigned int u32x4;
typedef __attribute__((ext_vector_type(8))) int i32x8;
typedef __attribute__((ext_vector_type(4))) int i32x4;

__device__ __forceinline__ v8f wmma_f32_4(v2f a, v2f b, v8f c) {
  // D(16x16,f32) += A(16x4,f32) * B(4x16,f32)
  return __builtin_amdgcn_wmma_f32_16x16x4_f32(false, a, false, b, (short)0, c,
                                               false, false);
}

// --------------------------- Tensor Data Mover: 2D tile global -> LDS ------
// D# layout per CDNA5 ISA ch.8 (group0: count/lds/global/type, group1:
// data_size + dims + tile dims + stride). Groups 2/3 zero (<=2D tensor).
// This toolchain uses the 6-arg builtin: (g0, g1, g2, g3, g_extra, cpol).
#if __has_builtin(__builtin_amdgcn_tensor_load_to_lds)
#define HAVE_TDM 1
__device__ __forceinline__ void tdm_load_2d(unsigned lds_off, const void* gptr,
                                            unsigned dim0, unsigned dim1,
                                            unsigned stride0) {
  unsigned long long ga = (unsigned long long)(size_t)gptr;
  u32x4 g0;
  g0[0] = 1u;                                            // count=1, user mode
  g0[1] = lds_off;                                       // lds_addr (bytes)
  g0[2] = (unsigned)ga;                                  // global_addr[31:0]
  g0[3] = (unsigned)((ga >> 32) & 0x01FFFFFFu) | (2u << 30);  // [56:32]+type=2
  i32x8 g1;
  g1[0] = (int)(2u << 16);                               // data_size=4 bytes
  g1[1] = (int)((dim0 & 0xFFFFu) << 16);                 // tensor_dim0[15:0]
  g1[2] = (int)((dim0 >> 16) | ((dim1 & 0xFFFFu) << 16));// dim0[31:16],dim1 lo
  g1[3] = (int)((dim1 >> 16) | ((dim0 & 0xFFFFu) << 16));// dim1 hi, tile0=dim0
  g1[4] = (int)(dim1 & 0xFFFFu);                         // tile1=dim1, tile2=0
  g1[5] = (int)stride0;                                  // dim0_stride[31:0]
  g1[6] = 0;                                             // stride hi bits = 0
  g1[7] = 0;
  i32x4 z4 = {0, 0, 0, 0};
  i32x8 z8 = {0, 0, 0, 0, 0, 0, 0, 0};
  __builtin_amdgcn_tensor_load_to_lds(g0, g1, z4, z4, z8, 0);
}
#else
#define HAVE_TDM 0
#endif

__device__ __forceinline__ unsigned lds_offset_of(const void* p) {
  return (unsigned)(size_t)p;  // addrspace(3) -> low 32 bits = LDS byte offset
}

// ---------------------------------------------------------------- utilities
__global__ void k_zero_f32(float* p, int n) {
  int i = blockIdx.x * 256 + threadIdx.x;
  if (i < n) p[i] = 0.f;
}
__global__ void k_zero_i32(int* p, int n) {
  int i = blockIdx.x * 256 + threadIdx.x;
  if (i < n) p[i] = 0;
}
__global__ void k_init_oi(int* oi) {
  int i = blockIdx.x * 256 + threadIdx.x;
  if (i < NMAX) oi[i] = i;
}
__global__ void k_init_edges(const int* __restrict__ ei, int* src, int* dst,
                             float* em) {
  int e = blockIdx.x * 256 + threadIdx.x;
  if (e >= EDG) return;
  src[e] = ei[e];
  dst[e] = ei[EDG + e];
  em[e] = 1.f;
}

// ------------------------------------------------- degree / score (integer)
__global__ void k_edge_deg(const int* __restrict__ src,
                           const int* __restrict__ dst,
                           const float* __restrict__ em, int* deg, int* score) {
  int e = blockIdx.x * 256 + threadIdx.x;
  if (e >= EDG) return;
  if (em[e] != 0.f) {
    atomicAdd(&deg[dst[e]], 1);
    atomicAdd(&score[src[e]], 1);
  }
}
__global__ void k_dinv(const int* __restrict__ deg, float* dinv, int n) {
  int i = blockIdx.x * 256 + threadIdx.x;
  if (i < n) dinv[i] = rsqrtf((float)deg[i] + 1.0f);
}

// ------------------------------------------- al/ar projection via f32 WMMA
// One wave per 16-node tile. A = xc rows (16 x 128, K-chunks of 4);
// B column 0 = att_l chunk, column 1 = att_r chunk, columns 2..15 = 0.
// att vectors staged in LDS; padding columns produced by cndmask (no exec
// branches around the WMMA feed loads).
__global__ void k_alar(const float* __restrict__ xc,
                       const float* __restrict__ alw,
                       const float* __restrict__ arw, float* al, float* ar,
                       int n) {
  __shared__ float s_att[2 * CDIM];
  {
    int t = threadIdx.x;            // 256 threads load 256 floats
    s_att[t] = (t < CDIM) ? alw[t] : arw[t - CDIM];
  }
  __syncthreads();
  int wid = (blockIdx.x * blockDim.x + threadIdx.x) >> 5;
  int lane = threadIdx.x & 31;
  int row0 = wid << 4;              // n is a multiple of 128 per block
  int m = lane & 15;                // A: row; B: column; D: column
  int kh = (lane >> 4) << 1;        // K-half select (0 or 2)
  const float* bbase = s_att + (m & 1) * CDIM;  // clamped, always valid
  bool bvalid = (m < 2);
  v8f c = {};
  for (int k = 0; k < CDIM; k += 4) {
    v2f a = *(const v2f*)(xc + (size_t)(row0 + m) * CDIM + k + kh);
    v2f t = *(const v2f*)(bbase + k + kh);      // unconditional ds_load_b64
    v2f b;
    b.x = bvalid ? t.x : 0.f;                   // v_cndmask, no branch
    b.y = bvalid ? t.y : 0.f;
    c = wmma_f32_4(a, b, c);
  }
  int mbase = (lane >> 4) * 8;  // D: lanes 0-15 -> M=vgpr, 16-31 -> M=vgpr+8
  if (m == 0) { for (int r = 0; r < 8; ++r) al[row0 + mbase + r] = c[r]; }
  else if (m == 1) { for (int r = 0; r < 8; ++r) ar[row0 + mbase + r] = c[r]; }
}

// ------------------------------------------------- edge messages (scatter)
__global__ void k_edge_msg(const int* __restrict__ src,
                           const int* __restrict__ dst,
                           const float* __restrict__ em,
                           const float* __restrict__ dinv,
                           const float* __restrict__ al,
                           const float* __restrict__ ar,
                           const float* __restrict__ xc, float* hacc) {
  int wid = (blockIdx.x * 256 + threadIdx.x) >> 5;  // one wave per edge
  if (wid >= EDG) return;
  int lane = threadIdx.x & 31;
  float m = em[wid];
  if (m == 0.f) return;
  int s = src[wid], d = dst[wid];
  float coef = dinv[s] * dinv[d] * m * tanhf(al[d] + ar[s]);
  float4 v = *(const float4*)(xc + (size_t)s * CDIM + lane * 4);
  float* hd = hacc + (size_t)d * CDIM + lane * 4;
  atomicAdd(hd + 0, coef * v.x);
  atomicAdd(hd + 1, coef * v.y);
  atomicAdd(hd + 2, coef * v.z);
  atomicAdd(hd + 3, coef * v.w);
}

// --------------------------------------- self-loop + eps residual + ReLU
__global__ void k_self_relu(const float* __restrict__ xc,
                            const float* __restrict__ x0,
                            const int* __restrict__ oi,
                            const float* __restrict__ dinv,
                            const float* __restrict__ al,
                            const float* __restrict__ ar, float* h, int n) {
  int wid = (blockIdx.x * 256 + threadIdx.x) >> 5;  // one wave per node
  if (wid >= n) return;
  int lane = threadIdx.x & 31;
  float di = dinv[wid];
  float coef = di * di * tanhf(al[wid] + ar[wid]);
  size_t o = (size_t)wid * CDIM + lane * 4;
  float4 v = *(const float4*)(xc + o);
  float4 xv = *(const float4*)(x0 + (size_t)oi[wid] * CDIM + lane * 4);
  float4 a = *(float4*)(h + o);
  float4 r;
  r.x = fmaxf(a.x + coef * v.x + EPS_FA * xv.x, 0.f);
  r.y = fmaxf(a.y + coef * v.y + EPS_FA * xv.y, 0.f);
  r.z = fmaxf(a.z + coef * v.z + EPS_FA * xv.z, 0.f);
  r.w = fmaxf(a.w + coef * v.w + EPS_FA * xv.w, 0.f);
  *(float4*)(h + o) = r;
}

// ------------------------------- global max pool (h >= 0 -> uint atomicMax)
__global__ void k_pool(const float* __restrict__ h, float* fused, int npg,
                       int layer) {
  int nb = blockIdx.x * 256;       // 256 consecutive nodes (never straddles G)
  int g = nb / npg;
  int q = threadIdx.x & 31;        // channel quad
  int rg = threadIdx.x >> 5;       // row group 0..7
  float4 vm = {0.f, 0.f, 0.f, 0.f};
  for (int i = rg; i < 256; i += 8) {
    float4 v = *(const float4*)(h + (size_t)(nb + i) * CDIM + q * 4);
    vm.x = fmaxf(vm.x, v.x); vm.y = fmaxf(vm.y, v.y);
    vm.z = fmaxf(vm.z, v.z); vm.w = fmaxf(vm.w, v.w);
  }
  unsigned* dp = (unsigned*)(fused + g * (CDIM * 3) + layer * CDIM + q * 4);
  atomicMax(dp + 0, __float_as_uint(vm.x));
  atomicMax(dp + 1, __float_as_uint(vm.y));
  atomicMax(dp + 2, __float_as_uint(vm.z));
  atomicMax(dp + 3, __float_as_uint(vm.w));
}

// --------------- per-graph top-k with exact (score desc, index asc) ranking
__global__ void k_topk(const int* __restrict__ score, int* nmap, int* perm,
                       int npg, int k) {
  __shared__ int hist[MAXS];
  __shared__ int sh_t;
  int g = blockIdx.x;
  int gb = g * npg;
  for (int v = threadIdx.x; v < MAXS; v += 256) hist[v] = 0;
  __syncthreads();
  for (int i = threadIdx.x; i < npg; i += 256) {
    int s = score[gb + i]; if (s > MAXS - 1) s = MAXS - 1;
    atomicAdd(&hist[s], 1);
  }
  __syncthreads();
  if (threadIdx.x == 0) {
    int run = 0, t = 0;
    for (int v = MAXS - 1; v >= 0; --v) {
      int hv = hist[v];
      hist[v] = run;           // becomes rank base: count of scores > v
      run += hv;
      if (run >= k) { t = v; break; }
    }
    sh_t = t;
  }
  __syncthreads();
  if (threadIdx.x >= 32) return;   // single wave does ordered assignment
  int lane = threadIdx.x;
  int t = sh_t;
  for (int base = 0; base < npg; base += 32) {
    int node = gb + base + lane;
    int s = score[node]; if (s > MAXS - 1) s = MAXS - 1;
    bool act = (s >= t);
    int cnt = 0, total = 0, lead = lane;
    for (int j = 0; j < 32; ++j) {
      int sj = __shfl(s, j, 32);
      int aj = __shfl((int)act, j, 32);
      if (act && aj && sj == s) {
        total++;
        if (j < lane) { cnt++; if (j < lead) lead = j; }
      }
    }
    int b = 0;
    if (act && cnt == 0) b = atomicAdd(&hist[s], total);  // chunk base
    b = __shfl(b, lead, 32);
    int nm = -1;
    if (act) {
      int rank = b + cnt;
      if (rank < k) { nm = g * k + rank; perm[nm] = node; }
    }
    nmap[node] = nm;
  }
}

__global__ void k_remap(int* src, int* dst, float* em,
                        const int* __restrict__ nmap) {
  int e = blockIdx.x * 256 + threadIdx.x;
  if (e >= EDG) return;
  int ns = nmap[src[e]], nd = nmap[dst[e]];
  float m = em[e];
  if (ns < 0 || nd < 0) m = 0.f;
  src[e] = ns < 0 ? 0 : ns;
  dst[e] = nd < 0 ? 0 : nd;
  em[e] = m;
}

__global__ void k_gather(const float* __restrict__ h,
                         const int* __restrict__ perm,
                         const int* __restrict__ oio, float* cur, int* oin,
                         int nnew) {
  int wid = (blockIdx.x * 256 + threadIdx.x) >> 5;
  if (wid >= nnew) return;
  int lane = threadIdx.x & 31;
  int old = perm[wid];
  *(float4*)(cur + (size_t)wid * CDIM + lane * 4) =
      *(const float4*)(h + (size_t)old * CDIM + lane * 4);
  if (lane == 0) oin[wid] = oio[old];
}

// ------------------------------------------- gi = fused @ w_ih^T  (WMMA)
// M=16 (rows 8..15 padded zero), N=768, K=384. 8 waves x 6 N-tiles.
// A (fused, 8x384) staged into LDS by the Tensor Data Mover.
__global__ void k_gi(const float* __restrict__ fused,
                     const float* __restrict__ w_ih,
                     const float* __restrict__ b_ih, float* gi) {
  __shared__ float sA[GCNT * 384];
#if HAVE_TDM
  if (threadIdx.x < 32) {
    tdm_load_2d(lds_offset_of(sA), fused, 384, GCNT, 384);
    __builtin_amdgcn_s_wait_tensorcnt(0);
  }
#else
  for (int i = threadIdx.x; i < GCNT * 384; i += 256) sA[i] = fused[i];
#endif
  __syncthreads();
  int wv = threadIdx.x >> 5, lane = threadIdx.x & 31;
  int m = lane & 15, kh = (lane >> 4) << 1;
  const float* abase = sA + (m & 7) * 384;   // clamped row, always valid
  bool avalid = (m < 8);
  for (int nt = wv; nt < 48; nt += 8) {
    int jb = nt * 16;
    v8f c = {};
    for (int k = 0; k < 384; k += 4) {
      v2f t = *(const v2f*)(abase + k + kh);  // unconditional ds_load_b64
      v2f a;
      a.x = avalid ? t.x : 0.f;
      a.y = avalid ? t.y : 0.f;
      v2f b = *(const v2f*)(w_ih + (size_t)(jb + m) * 384 + k + kh);
      c = wmma_f32_4(a, b, c);
    }
    if (lane < 16) {
      for (int r = 0; r < 8; ++r)
        gi[r * 768 + jb + lane] = c[r] + b_ih[jb + lane];
    }
  }
}

__global__ void k_gru(const float* __restrict__ gi,
                      const float* __restrict__ b_hh, float* hgru) {
  int i = blockIdx.x * 256 + threadIdx.x;
  if (i >= GCNT * HDIM) return;
  int g = i / HDIM, j = i % HDIM;
  float ir = gi[g * 768 + j] + b_hh[j];
  float iz = gi[g * 768 + 256 + j] + b_hh[256 + j];
  float inn = gi[g * 768 + 512 + j];
  float r = 1.f / (1.f + expf(-ir));
  float z = 1.f / (1.f + expf(-iz));
  float nc = tanhf(inn + r * b_hh[512 + j]);
  hgru[g * HDIM + j] = (1.f - z) * nc;
}

// ----- out = a*(hgru@w_fin^T + b_fin) + (1-a)*(root@w_root^T + b_root) (WMMA)
// hgru (8x256) and root (8x128) staged into LDS by the TDM.
__global__ void k_final(const float* __restrict__ hgru,
                        const float* __restrict__ w_fin,
                        const float* __restrict__ b_fin,
                        const float* __restrict__ root,
                        const float* __restrict__ w_root,
                        const float* __restrict__ b_root,
                        const float* __restrict__ alpha_p, float* out) {
  __shared__ float sH[GCNT * HDIM + GCNT * CDIM];
#if HAVE_TDM
  if (threadIdx.x < 32) {
    tdm_load_2d(lds_offset_of(sH), hgru, HDIM, GCNT, HDIM);
    tdm_load_2d(lds_offset_of(sH) + GCNT * HDIM * 4, root, CDIM, GCNT, CDIM);
    __builtin_amdgcn_s_wait_tensorcnt(0);
  }
#else
  for (int i = threadIdx.x; i < GCNT * HDIM; i += 256) sH[i] = hgru[i];
  for (int i = threadIdx.x; i < GCNT * CDIM; i += 256)
    sH[GCNT * HDIM + i] = root[i];
#endif
  __syncthreads();
  int wv = threadIdx.x >> 5, lane = threadIdx.x & 31;
  int m = lane & 15, kh = (lane >> 4) << 1;
  float alpha = alpha_p[0], beta = 1.f - alpha;
  bool avalid = (m < 8);
  const float* aH = sH + (m & 7) * HDIM;
  const float* aR = sH + GCNT * HDIM + (m & 7) * CDIM;
  int jb = wv * 16;  // 8 waves cover N=128
  v8f c = {};
  for (int k = 0; k < HDIM; k += 4) {
    v2f t = *(const v2f*)(aH + k + kh);
    v2f a;
    a.x = avalid ? alpha * t.x : 0.f;
    a.y = avalid ? alpha * t.y : 0.f;
    v2f b = *(const v2f*)(w_fin + (size_t)(jb + m) * HDIM + k + kh);
    c = wmma_f32_4(a, b, c);
  }
  for (int k = 0; k < CDIM; k += 4) {
    v2f t = *(const v2f*)(aR + k + kh);
    v2f a;
    a.x = avalid ? beta * t.x : 0.f;
    a.y = avalid ? beta * t.y : 0.f;
    v2f b = *(const v2f*)(w_root + (size_t)(jb + m) * CDIM + k + kh);
    c = wmma_f32_4(a, b, c);
  }
  if (lane < 16) {
    for (int r = 0; r < 8; ++r)
      out[r * OUTD + jb + lane] =
          c[r] + alpha * b_fin[jb + lane] + beta * b_root[jb + lane];
  }
}

// =========================================================================
extern "C" void kernel_launch(void* const* d_in, const int* in_sizes, int n_in,
                              void* d_out, int out_size, void* d_ws,
                              size_t ws_size, hipStream_t stream) {
  (void)in_sizes; (void)n_in; (void)out_size; (void)ws_size;
  const float* x      = (const float*)d_in[0];
  const int*   ei     = (const int*)d_in[1];
  const float* root   = (const float*)d_in[3];
  const float* att_l  = (const float*)d_in[4];
  const float* att_r  = (const float*)d_in[5];
  const float* w_ih   = (const float*)d_in[6];
  const float* b_ih   = (const float*)d_in[8];
  const float* b_hh   = (const float*)d_in[9];
  const float* w_fin  = (const float*)d_in[10];
  const float* b_fin  = (const float*)d_in[11];
  const float* w_root = (const float*)d_in[12];
  const float* b_root = (const float*)d_in[13];
  const float* alpha  = (const float*)d_in[14];
  float* out = (float*)d_out;

  char* w = (char*)d_ws;
  auto take = [&](size_t bytes) {
    void* p = (void*)w;
    w += ((bytes + 255) & ~(size_t)255);
    return p;
  };
  float* bufA  = (float*)take((size_t)NMAX * CDIM * 4);
  float* bufB  = (float*)take((size_t)NMAX * CDIM * 4);
  int*   srcb  = (int*)take((size_t)EDG * 4);
  int*   dstb  = (int*)take((size_t)EDG * 4);
  float* emb   = (float*)take((size_t)EDG * 4);
  int*   deg   = (int*)take((size_t)NMAX * 4);
  int*   score = (int*)take((size_t)NMAX * 4);
  float* dinv  = (float*)take((size_t)NMAX * 4);
  float* al    = (float*)take((size_t)NMAX * 4);
  float* ar    = (float*)take((size_t)NMAX * 4);
  int*   nmap  = (int*)take((size_t)NMAX * 4);
  int*   perm  = (int*)take((size_t)NMAX * 4);
  int*   oiA   = (int*)take((size_t)NMAX * 4);
  int*   oiB   = (int*)take((size_t)NMAX * 4);
  float* fused = (float*)take((size_t)GCNT * CDIM * 3 * 4);
  float* gi    = (float*)take((size_t)GCNT * 768 * 4);
  float* hgru  = (float*)take((size_t)GCNT * HDIM * 4);

  k_init_edges<<<EDG / 256, 256, 0, stream>>>(ei, srcb, dstb, emb);
  k_init_oi<<<NMAX / 256, 256, 0, stream>>>(oiA);
  k_zero_f32<<<(GCNT * CDIM * 3 + 255) / 256, 256, 0, stream>>>(
      fused, GCNT * CDIM * 3);

  int npg = NPG0;
  const float* cur = x;
  int* oic = oiA;
  int* oin = oiB;
  for (int L = 0; L < 3; ++L) {
    int n = GCNT * npg;
    int k = npg / 2;  // ceil(npg*0.5) for even npg
    k_zero_i32<<<n / 256, 256, 0, stream>>>(deg, n);
    k_zero_i32<<<n / 256, 256, 0, stream>>>(score, n);
    k_zero_f32<<<(n * CDIM) / 256, 256, 0, stream>>>(bufB, n * CDIM);
    k_edge_deg<<<EDG / 256, 256, 0, stream>>>(srcb, dstb, emb, deg, score);
    k_dinv<<<n / 256, 256, 0, stream>>>(deg, dinv, n);
    k_alar<<<n / 128, 256, 0, stream>>>(cur, att_l + L * CDIM,
                                        att_r + L * CDIM, al, ar, n);
    k_edge_msg<<<EDG / 8, 256, 0, stream>>>(srcb, dstb, emb, dinv, al, ar, cur,
                                            bufB);
    k_self_relu<<<n / 8, 256, 0, stream>>>(cur, x, oic, dinv, al, ar, bufB, n);
    k_pool<<<n / 256, 256, 0, stream>>>(bufB, fused, npg, L);
    if (L < 2) {
      k_topk<<<GCNT, 256, 0, stream>>>(score, nmap, perm, npg, k);
      k_remap<<<EDG / 256, 256, 0, stream>>>(srcb, dstb, emb, nmap);
      k_gather<<<(GCNT * k) / 8, 256, 0, stream>>>(bufB, perm, oic, bufA, oin,
                                                   GCNT * k);
      cur = bufA;
      int* t = oic; oic = oin; oin = t;
    }
    npg = k;
  }
  k_gi<<<1, 256, 0, stream>>>(fused, w_ih, b_ih, gi);
  k_gru<<<(GCNT * HDIM + 255) / 256, 256, 0, stream>>>(gi, b_hh, hgru);
  k_final<<<1, 256, 0, stream>>>(hgru, w_fin, b_fin, root, w_root, b_root,
                                 alpha, out);
}